// GCN_55645596287499
// MI455X (gfx1250) — compile-verified
//
#include <hip/hip_runtime.h>

typedef float v2f __attribute__((ext_vector_type(2)));
typedef float v4f __attribute__((ext_vector_type(4)));
typedef float v8f __attribute__((ext_vector_type(8)));
typedef int   v4i __attribute__((vector_size(16)));

#define NNODE 4096
#define BATCH 16
#define FDIM  64
#define LDSA  68   // A-tile row stride (floats): rows 16B-aligned, conflict-free frag reads
#define LDST  66   // transposed-tile row stride: even (v2f 8B-aligned), good bank spread

// ---- gfx1250 async global->LDS path (probe via __has_builtin; reg-staging fallback) ----
#if defined(__has_builtin)
#  if __has_builtin(__builtin_amdgcn_global_load_async_to_lds_b128)
#    define HAVE_ASYNC_LDS 1
#  endif
#endif
#ifndef HAVE_ASYNC_LDS
#  define HAVE_ASYNC_LDS 0
#endif

#if HAVE_ASYNC_LDS
typedef __attribute__((address_space(1))) v4i GV4;   // global b128
typedef __attribute__((address_space(3))) v4i LV4;   // LDS b128
__device__ __forceinline__ void async_tile_b128(const float* g, float* l) {
    __builtin_amdgcn_global_load_async_to_lds_b128((GV4*)g, (LV4*)l, 0, 0);
}
__device__ __forceinline__ void wait_asynccnt0() {
#  if defined(__has_builtin) && __has_builtin(__builtin_amdgcn_s_wait_asynccnt)
    __builtin_amdgcn_s_wait_asynccnt(0);
#  else
    asm volatile("s_wait_asynccnt 0x0" ::: "memory");
#  endif
}
#endif

// ---------------------------------------------------------------------------
// Kernel 0: d[n] = rsqrt(rowsum(adj[n,:]) with zero fixup)
// ---------------------------------------------------------------------------
__global__ __launch_bounds__(256) void gcn_degree_kernel(
    const float* __restrict__ adj, float* __restrict__ dinv)
{
    __shared__ float red[256];
    const int row = blockIdx.x;
    const float* a = adj + (size_t)row * NNODE;
    float s = 0.f;
    for (int c = threadIdx.x; c < NNODE; c += 256) s += a[c];
    red[threadIdx.x] = s;
    __syncthreads();
    for (int off = 128; off > 0; off >>= 1) {
        if (threadIdx.x < off) red[threadIdx.x] += red[threadIdx.x + off];
        __syncthreads();
    }
    if (threadIdx.x == 0) {
        float r = red[0];
        if (r == 0.f) r = 1.f;
        dinv[row] = rsqrtf(r);
    }
}

// ---------------------------------------------------------------------------
// Kernel 1: sup[b*N+m, :] = d[m] * (graph[b*N+m, :] @ W)
// ---------------------------------------------------------------------------
__global__ __launch_bounds__(256) void gcn_support_kernel(
    const float* __restrict__ graph,   // [B*N, 64]
    const float* __restrict__ weight,  // [64, 64]
    const float* __restrict__ dinv,    // [N]
    float* __restrict__ sup)           // [B*N, 64]
{
    __shared__ float Gs[64][LDSA];
    __shared__ float WsT[64][LDST];    // WsT[o][k] = W[k][o]
    const int tid  = threadIdx.x;
    const int row0 = blockIdx.x * 64;

    const v4f* gsrc = (const v4f*)(graph + (size_t)row0 * FDIM);
    const v4f* wsrc = (const v4f*)weight;
#pragma unroll
    for (int i = 0; i < 4; ++i) {
        const int idx = tid + 256 * i;
        const int r = idx >> 4, c = (idx & 15) * 4;
        *(v4f*)&Gs[r][c] = gsrc[idx];
        const v4f w = wsrc[idx];
#pragma unroll
        for (int j = 0; j < 4; ++j) WsT[c + j][r] = w[j];
    }
    __syncthreads();

    const int wave  = tid >> 5, lane = tid & 31;
    const int mblk  = (wave & 3) * 16;
    const int nblk  = (wave >> 2) * 32;
    const int lrow  = lane & 15;
    const int khalf = (lane >> 4) << 1;

    v8f acc0 = {}, acc1 = {};
#pragma unroll
    for (int kk = 0; kk < 64; kk += 4) {
        const int kb = kk + khalf;
        v2f a  = *(const v2f*)&Gs[mblk + lrow][kb];
        v2f b0 = *(const v2f*)&WsT[nblk + lrow][kb];
        v2f b1 = *(const v2f*)&WsT[nblk + 16 + lrow][kb];
        acc0 = __builtin_amdgcn_wmma_f32_16x16x4_f32(false, a, false, b0, (short)0, acc0, false, false);
        acc1 = __builtin_amdgcn_wmma_f32_16x16x4_f32(false, a, false, b1, (short)0, acc1, false, false);
    }

#pragma unroll
    for (int i = 0; i < 8; ++i) {
        const int row = row0 + mblk + i + ((lane >> 4) << 3);
        const float dm = dinv[row & (NNODE - 1)];
        const int col = nblk + lrow;
        sup[(size_t)row * FDIM + col]      = dm * acc0[i];
        sup[(size_t)row * FDIM + col + 16] = dm * acc1[i];
    }
}

// ---------------------------------------------------------------------------
// Kernel 2: out[b,n,:] = relu(d[n] * (adj[n,:] @ sup[b,:,:]) + bias)
// 128x64 block tile, 32x32 wave tiles, double-buffered LDS, async adj staging.
// ---------------------------------------------------------------------------
__global__ __launch_bounds__(256) void gcn_aggregate_kernel(
    const float* __restrict__ adj,    // [N, N]
    const float* __restrict__ sup,    // [B*N, 64]
    const float* __restrict__ dinv,   // [N]
    const float* __restrict__ bias,   // [64]
    float* __restrict__ out)          // [B*N, 64]
{
    __shared__ float As[2][128][LDSA];   // As[buf][m][k]
    __shared__ float SsT[2][64][LDST];   // SsT[buf][n][k] = sup_tile[k][n]
    const int tid  = threadIdx.x;
    const int row0 = blockIdx.x * 128;
    const int b    = blockIdx.y;

    const int wave  = tid >> 5, lane = tid & 31;
    const int mbase = (wave & 3) * 32;
    const int nbase = (wave >> 2) * 32;
    const int lrow  = lane & 15;
    const int khalf = (lane >> 4) << 1;

    v8f acc00 = {}, acc01 = {}, acc10 = {}, acc11 = {};

    const float* supb = sup + (size_t)b * NNODE * FDIM;

    // ---- prologue: stage K-tile 0 into buffer 0 ----
#if HAVE_ASYNC_LDS
#pragma unroll
    for (int i = 0; i < 8; ++i) {
        const int idx = tid + 256 * i;
        const int r = idx >> 4, c = (idx & 15) * 4;
        async_tile_b128(adj + (size_t)(row0 + r) * NNODE + c, &As[0][r][c]);
    }
#else
    {
        v4f ra[8];
#pragma unroll
        for (int i = 0; i < 8; ++i) {
            const int idx = tid + 256 * i;
            const int r = idx >> 4, c = (idx & 15) * 4;
            ra[i] = *(const v4f*)(adj + (size_t)(row0 + r) * NNODE + c);
        }
#pragma unroll
        for (int i = 0; i < 8; ++i) {
            const int idx = tid + 256 * i;
            const int r = idx >> 4, c = (idx & 15) * 4;
            *(v4f*)&As[0][r][c] = ra[i];
        }
    }
#endif
    {
        v4f rs[4];
#pragma unroll
        for (int i = 0; i < 4; ++i) {
            const int idx = tid + 256 * i;
            const int r = idx >> 4, c = (idx & 15) * 4;
            rs[i] = *(const v4f*)(supb + (size_t)r * FDIM + c);
        }
#pragma unroll
        for (int i = 0; i < 4; ++i) {
            const int idx = tid + 256 * i;
            const int r = idx >> 4, c = (idx & 15) * 4;
#pragma unroll
            for (int j = 0; j < 4; ++j) SsT[0][c + j][r] = rs[i][j];
        }
    }
#if HAVE_ASYNC_LDS
    wait_asynccnt0();
#endif
    __syncthreads();

    // ---- main loop: compute on buf p while filling buf 1-p ----
    for (int k0 = 0; k0 < NNODE; k0 += 64) {
        const int p  = (k0 >> 6) & 1;
        const int q  = 1 - p;
        const int kn = k0 + 64;

        v4f rnext[4];
#if !HAVE_ASYNC_LDS
        v4f ranext[8];
#endif
        if (kn < NNODE) {
#if HAVE_ASYNC_LDS
#pragma unroll
            for (int i = 0; i < 8; ++i) {
                const int idx = tid + 256 * i;
                const int r = idx >> 4, c = (idx & 15) * 4;
                async_tile_b128(adj + (size_t)(row0 + r) * NNODE + kn + c, &As[q][r][c]);
            }
#else
#pragma unroll
            for (int i = 0; i < 8; ++i) {
                const int idx = tid + 256 * i;
                const int r = idx >> 4, c = (idx & 15) * 4;
                ranext[i] = *(const v4f*)(adj + (size_t)(row0 + r) * NNODE + kn + c);
            }
#endif
#pragma unroll
            for (int i = 0; i < 4; ++i) {
                const int idx = tid + 256 * i;
                const int r = idx >> 4, c = (idx & 15) * 4;
                rnext[i] = *(const v4f*)(supb + (size_t)(kn + r) * FDIM + c);
            }
        }

#pragma unroll
        for (int kk = 0; kk < 64; kk += 4) {
            const int kb = kk + khalf;
            v2f a0 = *(const v2f*)&As[p][mbase + lrow][kb];
            v2f a1 = *(const v2f*)&As[p][mbase + 16 + lrow][kb];
            v2f b0 = *(const v2f*)&SsT[p][nbase + lrow][kb];
            v2f b1 = *(const v2f*)&SsT[p][nbase + 16 + lrow][kb];
            acc00 = __builtin_amdgcn_wmma_f32_16x16x4_f32(false, a0, false, b0, (short)0, acc00, false, false);
            acc01 = __builtin_amdgcn_wmma_f32_16x16x4_f32(false, a0, false, b1, (short)0, acc01, false, false);
            acc10 = __builtin_amdgcn_wmma_f32_16x16x4_f32(false, a1, false, b0, (short)0, acc10, false, false);
            acc11 = __builtin_amdgcn_wmma_f32_16x16x4_f32(false, a1, false, b1, (short)0, acc11, false, false);
        }

        if (kn < NNODE) {
#if !HAVE_ASYNC_LDS
#pragma unroll
            for (int i = 0; i < 8; ++i) {
                const int idx = tid + 256 * i;
                const int r = idx >> 4, c = (idx & 15) * 4;
                *(v4f*)&As[q][r][c] = ranext[i];
            }
#endif
#pragma unroll
            for (int i = 0; i < 4; ++i) {
                const int idx = tid + 256 * i;
                const int r = idx >> 4, c = (idx & 15) * 4;
#pragma unroll
                for (int j = 0; j < 4; ++j) SsT[q][c + j][r] = rnext[i][j];
            }
        }
#if HAVE_ASYNC_LDS
        wait_asynccnt0();
#endif
        __syncthreads();
    }

    // ---- epilogue: d[n]*acc + bias, relu ----
#pragma unroll
    for (int i = 0; i < 8; ++i) {
        const int rowA = row0 + mbase + i + ((lane >> 4) << 3);
        const int rowB = rowA + 16;
        const float dnA = dinv[rowA];
        const float dnB = dinv[rowB];
        const int col0 = nbase + lrow;
        const int col1 = col0 + 16;
        const float bs0 = bias[col0], bs1 = bias[col1];
        float v;
        v = dnA * acc00[i] + bs0; out[((size_t)b * NNODE + rowA) * FDIM + col0] = v > 0.f ? v : 0.f;
        v = dnA * acc01[i] + bs1; out[((size_t)b * NNODE + rowA) * FDIM + col1] = v > 0.f ? v : 0.f;
        v = dnB * acc10[i] + bs0; out[((size_t)b * NNODE + rowB) * FDIM + col0] = v > 0.f ? v : 0.f;
        v = dnB * acc11[i] + bs1; out[((size_t)b * NNODE + rowB) * FDIM + col1] = v > 0.f ? v : 0.f;
    }
}

// ---------------------------------------------------------------------------
extern "C" void kernel_launch(void* const* d_in, const int* in_sizes, int n_in,
                              void* d_out, int out_size, void* d_ws, size_t ws_size,
                              hipStream_t stream)
{
    const float* graph  = (const float*)d_in[0];  // [16,4096,64]
    const float* adj    = (const float*)d_in[1];  // [4096,4096]
    const float* weight = (const float*)d_in[2];  // [64,64]
    const float* bias   = (const float*)d_in[3];  // [64]
    float* out  = (float*)d_out;                  // output [16,4096,64] then adj [4096,4096]

    float* dinv = (float*)d_ws;                   // [4096]
    float* sup  = dinv + NNODE;                   // [16*4096*64]

    gcn_degree_kernel<<<NNODE, 256, 0, stream>>>(adj, dinv);

    gcn_support_kernel<<<(BATCH * NNODE) / 64, 256, 0, stream>>>(graph, weight, dinv, sup);

    dim3 grid(NNODE / 128, BATCH);
    gcn_aggregate_kernel<<<grid, 256, 0, stream>>>(adj, sup, dinv, bias, out);

    // Second tuple element: raw adj passthrough.
    (void)hipMemcpyAsync(out + (size_t)BATCH * NNODE * FDIM, adj,
                         (size_t)NNODE * NNODE * sizeof(float),
                         hipMemcpyDeviceToDevice, stream);
}